// ClassificationLoss_2173253452209
// MI455X (gfx1250) — compile-verified
//
#include <hip/hip_runtime.h>

typedef __attribute__((ext_vector_type(2))) float v2f;
typedef __attribute__((ext_vector_type(4))) float v4f;
typedef __attribute__((ext_vector_type(8))) float v8f;

// ds_swizzle group-of-32 encoding: offset = (xor<<10) | (or<<5) | and
#define SWZ(x, imm) __int_as_float(__builtin_amdgcn_ds_swizzle(__float_as_int(x), (imm)))
#define SWZ_X1  0x041F
#define SWZ_X2  0x081F
#define SWZ_X4  0x101F
#define SWZ_X8  0x201F
#define SWZ_X16 0x401F   // SWAPX16

#define NCLS 10
#define MIDC 5
#define EMB  1024
#define BATCH_N 32768
#define KPAIRS (EMB / 2)   // 512 K-pairs

__global__ __launch_bounds__(256) void loss_main_kernel(
    const float* __restrict__ reps, const float* __restrict__ W,
    const float* __restrict__ bias_g, const int* __restrict__ labels,
    float* __restrict__ partial)
{
    // W^T staged as K-pairs: wlds[k2 * NCLS + class] = {W[c][2k2], W[c][2k2+1]}
    __shared__ v2f   wlds[KPAIRS * NCLS];   // 40960 B
    __shared__ float blds[16];
    __shared__ float wsum[8];

    const int tid  = threadIdx.x;
    const int lane = tid & 31;
    const int wave = tid >> 5;
    const int n16  = lane & 15;     // class column (N) handled by this lane
    const int hi   = lane >> 4;     // 0 = lanes 0-15, 1 = lanes 16-31

    for (int idx = tid; idx < KPAIRS * NCLS; idx += 256) {
        int k2 = idx / NCLS, cn = idx % NCLS;
        v2f v;
        v.x = W[cn * EMB + 2 * k2];
        v.y = W[cn * EMB + 2 * k2 + 1];
        wlds[idx] = v;
    }
    if (tid < 16) blds[tid] = (tid < NCLS) ? bias_g[tid] : 0.0f;
    __syncthreads();

    const int   rowBase = blockIdx.x * 128 + wave * 16;
    const int   nclamp  = (n16 < NCLS) ? n16 : (NCLS - 1);  // lanes 10-15 read dup row (masked later)
    const float* arow   = reps + (size_t)(rowBase + n16) * EMB + 4 * hi;

    v8f acc = {0.f, 0.f, 0.f, 0.f, 0.f, 0.f, 0.f, 0.f};

    #pragma unroll 4
    for (int k = 0; k < EMB; k += 8) {
        v4f va = *(const v4f*)(arow + k);   // global_load_b128
        // cross-half exchange: low lanes need partner's {x,y}; high lanes need partner's {z,w}
        float t0 = hi ? va.x : va.z;
        float t1 = hi ? va.y : va.w;
        float s0 = SWZ(t0, SWZ_X16);
        float s1 = SWZ(t1, SWZ_X16);
        v2f a1, a2;
        a1.x = hi ? s0 : va.x;   a1.y = hi ? s1 : va.y;   // A frag, K = k..k+3
        a2.x = hi ? va.z : s0;   a2.y = hi ? va.w : s1;   // A frag, K = k+4..k+7
        const int kb = (k >> 1) + hi;
        v2f b1 = wlds[kb * NCLS + nclamp];        // ds_load_b64, conflict-free
        v2f b2 = wlds[(kb + 2) * NCLS + nclamp];
        acc = __builtin_amdgcn_wmma_f32_16x16x4_f32(false, a1, false, b1,
                                                    (short)0, acc, false, false);
        acc = __builtin_amdgcn_wmma_f32_16x16x4_f32(false, a2, false, b2,
                                                    (short)0, acc, false, false);
    }

    // Epilogue: D layout = VGPR j, lane -> row (j + 8*hi), col n16.
    const float bv    = blds[n16];
    const bool  valid = (n16 < NCLS);
    const int   rbase = rowBase + 8 * hi;
    float wave_tot = 0.0f;

    #pragma unroll
    for (int j = 0; j < 8; ++j) {
        float lg  = acc[j] + bv;
        int   lab = labels[rbase + j];
        // row max over valid classes (16-lane group reduction)
        float m = valid ? lg : -__builtin_inff();
        m = fmaxf(m, SWZ(m, SWZ_X1));
        m = fmaxf(m, SWZ(m, SWZ_X2));
        m = fmaxf(m, SWZ(m, SWZ_X4));
        m = fmaxf(m, SWZ(m, SWZ_X8));
        float e = valid ? __expf(lg - m) : 0.0f;
        float S = e;
        S += SWZ(S, SWZ_X1);
        S += SWZ(S, SWZ_X2);
        S += SWZ(S, SWZ_X4);
        S += SWZ(S, SWZ_X8);
        float s      = e / S;
        float invlog = __logf(1.0f - s + 1e-20f);
        bool  opp    = (lab < MIDC) ? (n16 >= MIDC) : (n16 < MIDC);
        float wgt    = (!valid || n16 == lab) ? 0.0f : (opp ? 2.0f : 1.0f);
        float r = wgt * invlog;
        r += SWZ(r, SWZ_X1);
        r += SWZ(r, SWZ_X2);
        r += SWZ(r, SWZ_X4);
        r += SWZ(r, SWZ_X8);
        wave_tot += r;   // r replicated within each 16-lane group = that row's loss
    }
    wave_tot += SWZ(wave_tot, SWZ_X16);   // combine low-half rows + high-half rows
    if (lane == 0) wsum[wave] = wave_tot;
    __syncthreads();
    if (tid == 0) {
        float t = 0.0f;
        #pragma unroll
        for (int i = 0; i < 8; ++i) t += wsum[i];
        partial[blockIdx.x] = t;
    }
}

__global__ __launch_bounds__(256) void loss_finalize_kernel(
    const float* __restrict__ partial, float* __restrict__ out)
{
    __shared__ float sm[256];
    int t = threadIdx.x;
    sm[t] = partial[t];
    __syncthreads();
    for (int s = 128; s > 0; s >>= 1) {
        if (t < s) sm[t] += sm[t + s];
        __syncthreads();
    }
    if (t == 0) out[0] = -sm[0] * (1.0f / (float)BATCH_N);
}

extern "C" void kernel_launch(void* const* d_in, const int* in_sizes, int n_in,
                              void* d_out, int out_size, void* d_ws, size_t ws_size,
                              hipStream_t stream) {
    const float* reps   = (const float*)d_in[0];
    const float* W      = (const float*)d_in[1];
    const float* b      = (const float*)d_in[2];
    const int*   labels = (const int*)d_in[3];
    float* partial = (float*)d_ws;   // 256 floats, fully overwritten each call

    loss_main_kernel<<<256, 256, 0, stream>>>(reps, W, b, labels, partial);
    loss_finalize_kernel<<<1, 256, 0, stream>>>(partial, (float*)d_out);
}